// TypeEnhancedGCNDeepseek0_10539849744796
// MI455X (gfx1250) — compile-verified
//
#include <hip/hip_runtime.h>
#include <hip/hip_bf16.h>

// ---- problem constants (from reference) ----
#define BB   8
#define LL   128
#define DIN  512
#define DOUT 512
#define DEPN 45
#define DEPD 16
#define LDW  (DIN + DEPD)   // 528, row stride of W1/W2
#define NROW (BB * LL)      // 1024

typedef __attribute__((ext_vector_type(2))) float v2f;
typedef __attribute__((ext_vector_type(8))) float v8f;

// fp32 WMMA: D = A(16x4) * B(4x16) + C(16x16), full-precision matrix pipe
__device__ __forceinline__ v8f wmma4(v2f a, v2f b, v8f c) {
    return __builtin_amdgcn_wmma_f32_16x16x4_f32(
        /*neg_a=*/false, a, /*neg_b=*/false, b,
        /*c_mod=*/(short)0, c, /*reuse_a=*/false, /*reuse_b=*/false);
}

// ---------- K0: per-batch aspect span (start/end) ----------
__global__ void k_startend(const unsigned char* __restrict__ mask, int* __restrict__ se) {
    int b = threadIdx.x;
    if (b >= BB) return;
    int first = -1, last = -1;
    for (int i = 0; i < LL; ++i) {
        if (mask[b * LL + i]) { if (first < 0) first = i; last = i; }
    }
    se[2 * b]     = (first < 0) ? 0 : first;
    se[2 * b + 1] = (last  < 0) ? (LL - 1) : last;
}

// ---------- K1: position-aware weighting, hw = h * w ----------
__global__ void k_hw(const float* __restrict__ h, const unsigned char* __restrict__ mask,
                     const int* __restrict__ se, float* __restrict__ hw) {
    int idx = blockIdx.x * blockDim.x + threadIdx.x;
    if (idx >= NROW * DIN) return;
    int row = idx / DIN;
    int b = row / LL, i = row % LL;
    int s = se[2 * b], e = se[2 * b + 1];
    const float invL = 1.0f / (float)LL;
    float w;
    if (mask[b * LL + i])      w = 0.0f;
    else if (i < s)            w = 1.0f - (float)(s - i) * invL;
    else if (i > e)            w = 1.0f - (float)(i - e) * invL;
    else                       w = 0.0f;
    hw[idx] = h[idx] * w;
}

// ---------- K2: F1[t,o] = emb[t,:]·W1[o,512:528], F2 likewise ----------
__global__ void k_F(const float* __restrict__ emb, const float* __restrict__ W1,
                    const float* __restrict__ W2, float* __restrict__ F1, float* __restrict__ F2) {
    int idx = blockIdx.x * blockDim.x + threadIdx.x;
    if (idx >= DEPN * DOUT) return;
    int t = idx / DOUT, o = idx % DOUT;
    float a1 = 0.f, a2 = 0.f;
    #pragma unroll
    for (int e = 0; e < DEPD; ++e) {
        float ev = emb[t * DEPD + e];
        a1 += ev * W1[o * LDW + DIN + e];
        a2 += ev * W2[o * LDW + DIN + e];
    }
    F1[idx] = a1;
    F2[idx] = a2;
}

// ---------- K3: c[t] = F1[t,:]·F2[t,:] ----------
__global__ void k_c(const float* __restrict__ F1, const float* __restrict__ F2,
                    float* __restrict__ cv) {
    int t = blockIdx.x * blockDim.x + threadIdx.x;
    if (t >= DEPN) return;
    float s = 0.f;
    for (int o = 0; o < DOUT; ++o) s += F1[t * DOUT + o] * F2[t * DOUT + o];
    cv[t] = s;
}

// ---------- K4: WMMA triple GEMM: U=hw@W1h^T+b1, V=hw@W2h^T+b2, HT=hw@Wg^T+bg ----------
// one wave per 16x16 tile; A-frag shared across the three B operands;
// 2-stage software pipeline so next step's loads overlap current step's WMMAs
__global__ __launch_bounds__(256) void k_gemm3(
    const float* __restrict__ A,
    const float* __restrict__ W1, const float* __restrict__ W2, const float* __restrict__ Wg,
    const float* __restrict__ b1, const float* __restrict__ b2, const float* __restrict__ bg,
    float* __restrict__ U, float* __restrict__ V, float* __restrict__ HT) {
    const int lane = threadIdx.x & 31;
    const int wave = threadIdx.x >> 5;
    const int tile = blockIdx.x * 8 + wave;          // 64 row-tiles * 32 col-tiles = 2048
    const int tr = tile >> 5;                        // /32
    const int tc = tile & 31;
    const int hh = lane >> 4;                        // half: 0 or 1
    const int ln = lane & 15;
    const int arow = tr * 16 + ln;                   // output row this lane feeds A for
    const int bcol = tc * 16 + ln;                   // output col (= weight row)

    const float* Ap  = A  + (size_t)arow * DIN + 2 * hh;
    const float* w1p = W1 + (size_t)bcol * LDW + 2 * hh;
    const float* w2p = W2 + (size_t)bcol * LDW + 2 * hh;
    const float* wgp = Wg + (size_t)bcol * DIN + 2 * hh;

    v8f accU = {}, accV = {}, accG = {};

    // pipeline stage 0
    v2f af  = *(const v2f*)(Ap);
    v2f bf1 = *(const v2f*)(w1p);
    v2f bf2 = *(const v2f*)(w2p);
    v2f bfg = *(const v2f*)(wgp);

    #pragma unroll 4
    for (int k = 0; k < DIN - 4; k += 4) {
        // issue next step's loads before this step's WMMAs
        v2f afn  = *(const v2f*)(Ap  + k + 4);
        v2f bf1n = *(const v2f*)(w1p + k + 4);
        v2f bf2n = *(const v2f*)(w2p + k + 4);
        v2f bfgn = *(const v2f*)(wgp + k + 4);
        accU = wmma4(af, bf1, accU);
        accV = wmma4(af, bf2, accV);
        accG = wmma4(af, bfg, accG);
        af = afn; bf1 = bf1n; bf2 = bf2n; bfg = bfgn;
    }
    accU = wmma4(af, bf1, accU);
    accV = wmma4(af, bf2, accV);
    accG = wmma4(af, bfg, accG);

    const float biasU = b1[bcol], biasV = b2[bcol], biasG = bg[bcol];
    #pragma unroll
    for (int r = 0; r < 8; ++r) {
        const int grow = tr * 16 + r + 8 * hh;       // C/D layout: M = r + 8*half
        const size_t o = (size_t)grow * DOUT + bcol; // N = lane%16
        U[o]  = accU[r] + biasU;
        V[o]  = accV[r] + biasV;
        HT[o] = accG[r] + biasG;
    }
}

// ---------- K5: logit table E[row,t] = (t==0)?0:exp(U·V + U·F2[t] + V·F1[t] + c[t]) ----------
__global__ __launch_bounds__(64) void k_logits(
    const float* __restrict__ U, const float* __restrict__ V,
    const float* __restrict__ F1, const float* __restrict__ F2,
    const float* __restrict__ cv, float* __restrict__ E) {
    __shared__ float uL[DIN], vL[DIN], red[64];
    const int row = blockIdx.x;
    const int t = threadIdx.x;
    for (int i = t; i < DIN; i += 64) {
        uL[i] = U[(size_t)row * DIN + i];
        vL[i] = V[(size_t)row * DIN + i];
    }
    __syncthreads();
    float p = 0.f;
    for (int i = t; i < DIN; i += 64) p += uL[i] * vL[i];
    red[t] = p;
    __syncthreads();
    for (int s2 = 32; s2 > 0; s2 >>= 1) {
        if (t < s2) red[t] += red[t + s2];
        __syncthreads();
    }
    const float S = red[0];
    if (t < DEPN) {
        if (t == 0) {
            E[(size_t)row * DEPN] = 0.0f;            // adjacency mask: dep==0 contributes nothing
        } else {
            const float* f1 = F1 + (size_t)t * DOUT;
            const float* f2 = F2 + (size_t)t * DOUT;
            float g = 0.f;
            for (int i = 0; i < DIN; ++i) g += uL[i] * f2[i] + vL[i] * f1[i];
            E[(size_t)row * DEPN + t] = expf(S + g + cv[t]);
        }
    }
}

// ---------- K6: inv[row] = 1 / (sum_j E[row, dep[row,j]] + 1e-6) ----------
__global__ void k_inv(const int* __restrict__ dep, const float* __restrict__ E,
                      float* __restrict__ inv) {
    int row = blockIdx.x * blockDim.x + threadIdx.x;
    if (row >= NROW) return;
    const int* dp = dep + (size_t)row * LL;
    const float* Er = E + (size_t)row * DEPN;
    float s = 0.f;
    for (int j = 0; j < LL; ++j) s += Er[dp[j]];
    inv[row] = 1.0f / (s + 1e-6f);
}

// ---------- K7: WMMA aggregation: out = relu( (E[dep] @ HT) * inv + bias ) ----------
// pipelined: dep-index load -> E gather -> WMMA chain overlapped across k-steps
__global__ __launch_bounds__(256) void k_agg(
    const int* __restrict__ dep, const float* __restrict__ E,
    const float* __restrict__ HT, const float* __restrict__ inv,
    const float* __restrict__ bias, float* __restrict__ out) {
    const int lane = threadIdx.x & 31;
    const int wave = threadIdx.x >> 5;
    const int tile = blockIdx.x * 8 + wave;          // 8 batches * 8 * 32 = 2048
    const int b  = tile >> 8;
    const int t2 = tile & 255;
    const int tr = t2 >> 5;                          // 0..7
    const int tc = t2 & 31;                          // 0..31
    const int hh = lane >> 4;
    const int ln = lane & 15;
    const int am = tr * 16 + ln;                     // i within batch
    const int gn = tc * 16 + ln;                     // output col

    const int*   dp = dep + ((size_t)(b * LL + am)) * LL + 2 * hh;
    const float* Er = E   + ((size_t)(b * LL + am)) * DEPN;
    const float* Bp = HT  + ((size_t)(b * LL + 2 * hh)) * DOUT + gn;

    v8f acc = {};

    // pipeline stage 0
    int2 tt = *(const int2*)(dp);
    v2f bf; bf.x = Bp[0]; bf.y = Bp[DOUT];
    float e0 = Er[tt.x], e1 = Er[tt.y];

    #pragma unroll 4
    for (int k = 0; k < LL - 4; k += 4) {
        int2 tn = *(const int2*)(dp + k + 4);
        v2f bn; bn.x = Bp[(size_t)(k + 4) * DOUT]; bn.y = Bp[(size_t)(k + 5) * DOUT];
        float e0n = Er[tn.x], e1n = Er[tn.y];       // gather for next step in flight
        v2f af; af.x = e0; af.y = e1;
        acc = wmma4(af, bf, acc);
        e0 = e0n; e1 = e1n; bf = bn;
    }
    {
        v2f af; af.x = e0; af.y = e1;
        acc = wmma4(af, bf, acc);
    }

    const float bb = bias[gn];
    #pragma unroll
    for (int r = 0; r < 8; ++r) {
        const int gi = tr * 16 + r + 8 * hh;
        float val = acc[r] * inv[b * LL + gi] + bb;
        out[((size_t)(b * LL + gi)) * DOUT + gn] = fmaxf(val, 0.0f);
    }
}

extern "C" void kernel_launch(void* const* d_in, const int* in_sizes, int n_in,
                              void* d_out, int out_size, void* d_ws, size_t ws_size,
                              hipStream_t stream) {
    const float*         h    = (const float*)d_in[0];
    const int*           dep  = (const int*)d_in[1];
    const unsigned char* mask = (const unsigned char*)d_in[2];   // bool -> 1 byte
    const float*         emb  = (const float*)d_in[3];
    const float*         W1   = (const float*)d_in[4];
    const float*         b1   = (const float*)d_in[5];
    const float*         W2   = (const float*)d_in[6];
    const float*         b2   = (const float*)d_in[7];
    const float*         Wg   = (const float*)d_in[8];
    const float*         bg   = (const float*)d_in[9];
    const float*         bias = (const float*)d_in[10];
    float* out = (float*)d_out;

    char* ws = (char*)d_ws;
    size_t off = 0;
    auto alloc = [&](size_t bytes) -> void* {
        void* p = ws + off;
        off = (off + bytes + 255) & ~(size_t)255;
        return p;
    };
    int*   se  = (int*)  alloc(BB * 2 * sizeof(int));
    float* hw  = (float*)alloc((size_t)NROW * DIN * sizeof(float));
    float* U   = (float*)alloc((size_t)NROW * DOUT * sizeof(float));
    float* V   = (float*)alloc((size_t)NROW * DOUT * sizeof(float));
    float* HT  = (float*)alloc((size_t)NROW * DOUT * sizeof(float));
    float* F1  = (float*)alloc((size_t)DEPN * DOUT * sizeof(float));
    float* F2  = (float*)alloc((size_t)DEPN * DOUT * sizeof(float));
    float* cv  = (float*)alloc((size_t)DEPN * sizeof(float));
    float* E   = (float*)alloc((size_t)NROW * DEPN * sizeof(float));
    float* inv = (float*)alloc((size_t)NROW * sizeof(float));
    (void)in_sizes; (void)n_in; (void)out_size; (void)ws_size;

    k_startend<<<1, 32, 0, stream>>>(mask, se);
    k_hw<<<(NROW * DIN) / 256, 256, 0, stream>>>(h, mask, se, hw);
    k_F<<<(DEPN * DOUT + 255) / 256, 256, 0, stream>>>(emb, W1, W2, F1, F2);
    k_c<<<1, 64, 0, stream>>>(F1, F2, cv);
    k_gemm3<<<(64 * 32) / 8, 256, 0, stream>>>(hw, W1, W2, Wg, b1, b2, bg, U, V, HT);
    k_logits<<<NROW, 64, 0, stream>>>(U, V, F1, F2, cv, E);
    k_inv<<<(NROW + 255) / 256, 256, 0, stream>>>(dep, E, inv);
    k_agg<<<(BB * 8 * 32) / 8, 256, 0, stream>>>(dep, E, HT, inv, bias, out);
}